// WordleModel_47278999994754
// MI455X (gfx1250) — compile-verified
//
#include <hip/hip_runtime.h>
#include <hip/hip_bf16.h>
#include <math.h>

typedef __attribute__((ext_vector_type(2))) float v2f;
typedef __attribute__((ext_vector_type(8))) float v8f;
typedef __attribute__((ext_vector_type(4))) unsigned int u32x4;
typedef __attribute__((ext_vector_type(4))) int i32x4;
typedef __attribute__((ext_vector_type(8))) int i32x8;

#define N_WORDS 100000
#define K_HID   1482           // 57 * 26
#define KTILE   32             // K rows per TDM stage
#define NSTAGE  (1472 / KTILE) // 46 stages cover k = 0..1471
#define KTAIL   1472

// TDM builtins exist only in the AMDGCN device pass; the x86 host pass must
// still parse this file, so it takes the (never-executed) fallback branch.
#if defined(__HIP_DEVICE_COMPILE__)
#  if defined(__has_builtin)
#    if __has_builtin(__builtin_amdgcn_tensor_load_to_lds) && __has_builtin(__builtin_amdgcn_s_wait_tensorcnt)
#      define USE_TDM 1
#    endif
#  endif
#  ifndef USE_TDM
#    error "CDNA5 device pass lacks __builtin_amdgcn_tensor_load_to_lds / s_wait_tensorcnt"
#  endif
#else
#  define USE_TDM 0
#endif

// ---------------------------------------------------------------------------
// Kernel 1: front end. 57 outputs: v(26) | inv(26) | cur(5). One block of 64.
// ---------------------------------------------------------------------------
__global__ void front_kernel(const float* __restrict__ xv, const float* __restrict__ xi,
                             const float* __restrict__ xc,
                             const float* __restrict__ Wv, const float* __restrict__ bv,
                             const float* __restrict__ Wi, const float* __restrict__ bi,
                             const float* __restrict__ Wc, const float* __restrict__ bc,
                             float* __restrict__ h57) {
    int t = threadIdx.x;
    if (t < 26) {
        float s = bv[t];
        #pragma unroll 13
        for (int k = 0; k < 26; ++k) s = fmaf(xv[k], Wv[k * 26 + t], s);
        h57[t] = s;
    } else if (t < 52) {
        int c = t - 26;
        float s = bi[c];
        #pragma unroll 13
        for (int k = 0; k < 26; ++k) s = fmaf(xi[k], Wi[k * 26 + c], s);
        h57[26 + c] = s;
    } else if (t < 57) {
        int c = t - 52;
        float s = bc[c];
        #pragma unroll
        for (int k = 0; k < 5; ++k) s = fmaf(xc[k], Wc[k * 5 + c], s);
        h57[52 + c] = s;
    }
}

// ---------------------------------------------------------------------------
// Kernel 2: h1482 = h57 @ W1 + b1.  W1 (57,1482) row-major -> coalesced cols.
// ---------------------------------------------------------------------------
__global__ void layer1_kernel(const float* __restrict__ h57, const float* __restrict__ W1,
                              const float* __restrict__ b1, float* __restrict__ h) {
    __shared__ float sh[57];
    if (threadIdx.x < 57) sh[threadIdx.x] = h57[threadIdx.x];
    __syncthreads();
    int o = blockIdx.x * blockDim.x + threadIdx.x;
    if (o < K_HID) {
        float s = b1[o];
        #pragma unroll 19
        for (int k = 0; k < 57; ++k) s = fmaf(sh[k], W1[k * K_HID + o], s);
        h[o] = s;
    }
}

// ---------------------------------------------------------------------------
// TDM descriptor: load a KTILE(=32 rows) x 128-col fp32 tile of W2 into LDS.
// Group0: count=1 | lds_addr | 57-bit global byte addr | type=2 ("image").
// Group1: data_size=4B, tensor_dim0=100000 (OOB cols read 0), tensor_dim1=1482,
//         tile_dim0=128, tile_dim1=32, tensor_dim0_stride=100000.
// Groups 2/3 zero (2-D tile).
// ---------------------------------------------------------------------------
#if USE_TDM
__device__ __forceinline__ void tdm_load_tile(const float* gsrc, unsigned lds_byte_addr) {
    unsigned long long ga = (unsigned long long)(uintptr_t)gsrc;
    u32x4 g0;
    g0[0] = 1u;                                            // count=1, user mode
    g0[1] = lds_byte_addr;                                 // lds_addr
    g0[2] = (unsigned)ga;                                  // global_addr[31:0]
    g0[3] = ((unsigned)(ga >> 32) & 0x01ffffffu) | (2u << 30); // addr[56:32] | type=2
    i32x8 g1;
    g1[0] = (int)0x00020000u;   // data_size=2 (4 bytes); mask/flags = 0
    g1[1] = (int)0x86A00000u;   // tensor_dim0[15:0]=0x86A0 in bits 63:48
    g1[2] = (int)0x05CA0001u;   // tensor_dim0[31:16]=1 | tensor_dim1[15:0]=1482
    g1[3] = (int)0x00800000u;   // tensor_dim1[31:16]=0 | tile_dim0=128
    g1[4] = (int)KTILE;         // tile_dim1=32 | tile_dim2=0
    g1[5] = (int)100000;        // tensor_dim0_stride[31:0]
    g1[6] = 0;                  // stride[47:32]=0 | dim1_stride[15:0]=0
    g1[7] = 0;
    i32x4 gz;  gz[0] = gz[1] = gz[2] = gz[3] = 0;
#if defined(__clang_major__) && __clang_major__ >= 23
    i32x8 gz8; gz8[0]=gz8[1]=gz8[2]=gz8[3]=gz8[4]=gz8[5]=gz8[6]=gz8[7]=0;
    __builtin_amdgcn_tensor_load_to_lds(g0, g1, gz, gz, gz8, 0);
#else
    __builtin_amdgcn_tensor_load_to_lds(g0, g1, gz, gz, 0);
#endif
}
#endif

// ---------------------------------------------------------------------------
// Kernel 3: logits = h(1482) @ W2(1482,100000) + b2 via V_WMMA_F32_16X16X4_F32.
// One wave per 16-column strip (8 waves / 128 cols per block). W2 staged
// through LDS by the Tensor Data Mover (16 KB tiles, double-buffered), paced
// with s_wait_tensorcnt + workgroup barriers. h broadcast into all 16 A-rows
// so every D row equals the logits; D VGPR0 lanes 0-15 give cols n0..n0+15.
// ---------------------------------------------------------------------------
__global__ void gemv_wmma_kernel(const float* __restrict__ h, const float* __restrict__ W2,
                                 const float* __restrict__ b2, float* __restrict__ logits) {
    __shared__ float sh[K_HID];
    __shared__ float tile[2][KTILE * 128];   // 2 x 16 KB

    for (int i = threadIdx.x; i < K_HID; i += blockDim.x) sh[i] = h[i];

    const int lane = threadIdx.x & 31;
    const int wave = threadIdx.x >> 5;
    const int nl   = lane & 15;
    const int n    = blockIdx.x * 128 + wave * 16 + nl;
    const int kh   = (lane >> 4) * 2;        // K sub-row pair: 0 or 2
    const int nblk = blockIdx.x * 128;       // block column base
    const int tcol = wave * 16 + nl;         // column within the LDS tile

    v8f acc = {};

#if USE_TDM
    if (wave == 0)
        tdm_load_tile(W2 + nblk, (unsigned)(uintptr_t)&tile[0][0]);

    for (int s = 0; s < NSTAGE; ++s) {
        const int cur = s & 1;
        if (wave == 0) {
            if (s + 1 < NSTAGE) {
                tdm_load_tile(W2 + (size_t)(s + 1) * KTILE * N_WORDS + nblk,
                              (unsigned)(uintptr_t)&tile[cur ^ 1][0]);
                __builtin_amdgcn_s_wait_tensorcnt(1);   // stage s complete
            } else {
                __builtin_amdgcn_s_wait_tensorcnt(0);
            }
        }
        __syncthreads();                     // tile[cur] (and sh) visible to all
        const float* __restrict__ t = &tile[cur][0];
        const int kb = s * KTILE;
        #pragma unroll
        for (int j = 0; j < KTILE / 4; ++j) {
            const int kk = j * 4 + kh;
            v2f a, b;
            a.x = sh[kb + kk];
            a.y = sh[kb + kk + 1];
            b.x = t[kk * 128 + tcol];
            b.y = t[(kk + 1) * 128 + tcol];
            acc = __builtin_amdgcn_wmma_f32_16x16x4_f32(
                false, a, false, b, (short)0, acc, false, false);
        }
        __syncthreads();                     // all waves done before tile[cur] reload
    }
#else
    // Host-parse / non-TDM fallback: cooperative b32 tile loads (single buffer).
    for (int s = 0; s < NSTAGE; ++s) {
        __syncthreads();
        const int kb = s * KTILE;
        for (int e = threadIdx.x; e < KTILE * 128; e += blockDim.x) {
            int row = e >> 7, col = e & 127;
            int gcol = nblk + col;
            tile[0][e] = (gcol < N_WORDS) ? W2[(size_t)(kb + row) * N_WORDS + gcol] : 0.0f;
        }
        __syncthreads();
        const float* __restrict__ t = &tile[0][0];
        #pragma unroll
        for (int j = 0; j < KTILE / 4; ++j) {
            const int kk = j * 4 + kh;
            v2f a, b;
            a.x = sh[kb + kk];
            a.y = sh[kb + kk + 1];
            b.x = t[kk * 128 + tcol];
            b.y = t[(kk + 1) * 128 + tcol];
            acc = __builtin_amdgcn_wmma_f32_16x16x4_f32(
                false, a, false, b, (short)0, acc, false, false);
        }
    }
#endif

    // Tail rows 1472..1481 + bias; D row M=0 lives in lanes 0-15 of acc[0].
    if (lane < 16 && n < N_WORDS) {
        float r = acc[0];
        #pragma unroll
        for (int k = KTAIL; k < K_HID; ++k)
            r = fmaf(sh[k], W2[(size_t)k * N_WORDS + n], r);
        r += b2[n];
        logits[n] = r;
    }
}

// ---------------------------------------------------------------------------
// Kernel 4: softmax statistics (global max, sum of exp). One 1024-thread block.
// ---------------------------------------------------------------------------
__global__ void softmax_stats_kernel(const float* __restrict__ logits, float* __restrict__ stats) {
    __shared__ float red[1024];
    const int t = threadIdx.x;

    float m = -INFINITY;
    for (int i = t; i < N_WORDS; i += 1024) m = fmaxf(m, logits[i]);
    red[t] = m;
    __syncthreads();
    for (int s = 512; s > 0; s >>= 1) {
        if (t < s) red[t] = fmaxf(red[t], red[t + s]);
        __syncthreads();
    }
    const float gmax = red[0];
    __syncthreads();

    float sum = 0.0f;
    for (int i = t; i < N_WORDS; i += 1024) sum += __expf(logits[i] - gmax);
    red[t] = sum;
    __syncthreads();
    for (int s = 512; s > 0; s >>= 1) {
        if (t < s) red[t] += red[t + s];
        __syncthreads();
    }
    if (t == 0) { stats[0] = gmax; stats[1] = red[0]; }
}

// ---------------------------------------------------------------------------
// Kernel 5: p = exp(l - max)/sum, masked by inv-letter hits and slot mismatch.
// In-place over d_out (each thread reads/writes only its own element).
// ---------------------------------------------------------------------------
__global__ void finalize_kernel(const float* __restrict__ stats,
                                const float* __restrict__ inv,     // h57 + 26
                                const float* __restrict__ xcur,    // raw x_current (5)
                                const int*   __restrict__ words,   // (N_WORDS, 5)
                                float* __restrict__ out) {
    int nIdx = blockIdx.x * blockDim.x + threadIdx.x;
    if (nIdx >= N_WORDS) return;

    const float gmax = stats[0];
    const float inv_sum = 1.0f / stats[1];
    float p = __expf(out[nIdx] - gmax) * inv_sum;

    bool kill = false;
    #pragma unroll
    for (int j = 0; j < 5; ++j) {
        int w = words[nIdx * 5 + j];
        if (inv[w] == 1.0f) kill = true;
        float c = xcur[j];
        if (c != -1.0f && c != (float)w) kill = true;
    }
    out[nIdx] = kill ? 0.0f : p;
}

// ---------------------------------------------------------------------------
// Launch
// ---------------------------------------------------------------------------
extern "C" void kernel_launch(void* const* d_in, const int* in_sizes, int n_in,
                              void* d_out, int out_size, void* d_ws, size_t ws_size,
                              hipStream_t stream) {
    const float* xv    = (const float*)d_in[0];   // x_valid   (26)
    const float* xi    = (const float*)d_in[1];   // x_invalid (26)
    const float* xc    = (const float*)d_in[2];   // x_current (5)
    // d_in[3] = x_turns (unused, as in reference)
    const int*   words = (const int*)  d_in[4];   // (100000, 5)
    const float* Wv    = (const float*)d_in[5];
    const float* bv    = (const float*)d_in[6];
    const float* Wi    = (const float*)d_in[7];
    const float* bi    = (const float*)d_in[8];
    const float* Wc    = (const float*)d_in[9];
    const float* bc    = (const float*)d_in[10];
    // d_in[11], d_in[12] = Wt, bt (unused)
    const float* W1    = (const float*)d_in[13];
    const float* b1    = (const float*)d_in[14];
    const float* W2    = (const float*)d_in[15];
    const float* b2    = (const float*)d_in[16];

    float* out   = (float*)d_out;                 // logits, then final p (in place)
    float* wsf   = (float*)d_ws;
    float* h57   = wsf;                           // [0, 57)
    float* h     = wsf + 64;                      // [64, 64+1482)
    float* stats = wsf + 1600;                    // [1600, 1602)

    front_kernel<<<1, 64, 0, stream>>>(xv, xi, xc, Wv, bv, Wi, bi, Wc, bc, h57);

    layer1_kernel<<<(K_HID + 255) / 256, 256, 0, stream>>>(h57, W1, b1, h);

    // 8 waves/block, 16 columns per wave -> 128 columns per block
    gemv_wmma_kernel<<<(N_WORDS + 127) / 128, 256, 0, stream>>>(h, W2, b2, out);

    softmax_stats_kernel<<<1, 1024, 0, stream>>>(out, stats);

    finalize_kernel<<<(N_WORDS + 255) / 256, 256, 0, stream>>>(stats, h57 + 26, xc, words, out);
}